// mySTAEformer_78477642432780
// MI455X (gfx1250) — compile-verified
//
#include <hip/hip_runtime.h>

// ---------------------------------------------------------------------------
// Problem constants (from reference): B=4, S=12, N=512, D=128, H=8, HD=16
// ---------------------------------------------------------------------------
#define BB 4
#define SS 12
#define NN 512
#define DD 128
#define HH 8
#define HD 16

__constant__ float QLEV[19] = {
    0.01f, 0.1f, 0.11f, 0.2f, 0.21f, 0.3f, 0.31f, 0.4f, 0.41f,
    0.5f,  0.51f, 0.6f, 0.61f, 0.7f, 0.71f, 0.8f, 0.81f, 0.9f, 0.91f};

#define CONGEST (-0.2260138304488262f)
#define NEG_BIG (-1.0e30f)

typedef float v2f __attribute__((ext_vector_type(2)));
typedef float v8f __attribute__((ext_vector_type(8)));

// D(16x16,f32) = A(16x4,f32) x B(4x16,f32) + C
// Layouts (ISA 7.12.2):
//  A: lane l -> M = l%16, VGPR v -> K = v + 2*(l/16)
//  B: lane l -> N = l%16, VGPR v -> K = v + 2*(l/16)   (mirror of A)
//  C/D: lane l -> N = l%16, VGPR r -> M = r + 8*(l/16)
__device__ __forceinline__ v8f wmma4(v2f a, v2f b, v8f c) {
  return __builtin_amdgcn_wmma_f32_16x16x4_f32(
      /*neg_a=*/false, a, /*neg_b=*/false, b,
      /*c_mod=*/(short)0, c, /*reuse_a=*/false, /*reuse_b=*/false);
}

// ---------------------------------------------------------------------------
// Kernel 1: LSH mask -> additive bias (0 or -1e30), one block per (b, i) row.
// ---------------------------------------------------------------------------
__global__ __launch_bounds__(256) void lsh_mask_kernel(
    const float* __restrict__ x, float* __restrict__ bias) {
  __shared__ float xi[SS];
  __shared__ float dist[NN];
  __shared__ float sorted[NN];
  __shared__ float xmean[NN];
  __shared__ float qv[19];

  const int bi = blockIdx.x;       // b*N + i
  const int b  = bi >> 9;
  const int i  = bi & (NN - 1);
  const int t  = threadIdx.x;
  const float* xb = x + (size_t)b * NN * SS;

  if (t < SS) xi[t] = xb[(size_t)i * SS + t];
  __syncthreads();

  for (int j = t; j < NN; j += 256) {
    float d = 0.0f, m = 0.0f;
    #pragma unroll
    for (int s = 0; s < SS; ++s) {
      float xv = xb[(size_t)j * SS + s];
      d += fabsf(xi[s] - xv);
      m += xv;
    }
    dist[j]   = d;
    sorted[j] = d;
    xmean[j]  = m * (1.0f / SS);
  }
  __syncthreads();

  // Bitonic sort of 512 floats with 256 threads (one pair per thread/step).
  for (int k = 2; k <= NN; k <<= 1) {
    for (int j = k >> 1; j > 0; j >>= 1) {
      int idx = ((t & ~(j - 1)) << 1) | (t & (j - 1));
      int ixj = idx | j;
      float a = sorted[idx];
      float c = sorted[ixj];
      bool up = ((idx & k) == 0);
      if ((a > c) == up) { sorted[idx] = c; sorted[ixj] = a; }
      __syncthreads();
    }
  }

  if (t < 19) {
    float pos = QLEV[t] * (float)(NN - 1);
    int lo = (int)pos;
    float fr = pos - (float)lo;
    qv[t] = sorted[lo] * (1.0f - fr) + sorted[lo + 1] * fr;
  }
  __syncthreads();

  float* brow = bias + (size_t)bi * NN;
  for (int j = t; j < NN; j += 256) {
    float d = dist[j];
    bool m = (d <= qv[0]) || (xmean[j] <= CONGEST);
    #pragma unroll
    for (int kq = 0; kq < 9; ++kq)
      m = m || (d >= qv[2 * kq + 1] && d <= qv[2 * kq + 2]);
    brow[j] = m ? 0.0f : NEG_BIG;
  }
}

// ---------------------------------------------------------------------------
// Kernel 2: Y = X @ W^T + bias, X is (rows x 128), W is (128 x 128) row-major.
// Block: 256 threads = 8 waves; each block computes a 128x128 output tile.
// K staged in panels of 32 through LDS. B[k][n] = W[n][k], so a row-major W
// panel makes every A and B fragment a contiguous, 8B-aligned ds_load_b64.
// ---------------------------------------------------------------------------
#define XSTR 36   // multiple of 4 (float4 staging) ; 36*m mod 64 distinct, m<16

__global__ __launch_bounds__(256) void proj_kernel(
    const float* __restrict__ X, const float* __restrict__ W,
    const float* __restrict__ bias, float* __restrict__ Y) {
  __shared__ float Xs[128 * XSTR];   // 18 KB
  __shared__ float Ws[128 * XSTR];   // 18 KB  (row-major W panel: Ws[n][k])

  const int t    = threadIdx.x;
  const int lane = t & 31;
  const int w    = t >> 5;
  const int hl   = lane >> 4;   // half-wave select
  const int lm   = lane & 15;
  const int rbase = blockIdx.x * 128;
  const int m0 = w * 16;

  v8f acc[8];
  #pragma unroll
  for (int nt = 0; nt < 8; ++nt) {
    float bv = bias[nt * 16 + lm];
    #pragma unroll
    for (int r = 0; r < 8; ++r) acc[nt][r] = bv;
  }

  for (int kp = 0; kp < 128; kp += 32) {
    __syncthreads();
    // Stage X panel: 128 rows x 32 cols (1024 float4 / 256 threads).
    #pragma unroll
    for (int it = 0; it < 4; ++it) {
      int f4  = t + it * 256;
      int row = f4 >> 3;
      int c0  = (f4 & 7) << 2;
      const float4 v = *(const float4*)(X + (size_t)(rbase + row) * DD + kp + c0);
      *(float4*)&Xs[row * XSTR + c0] = v;
    }
    // Stage W panel (row-major, no transpose).
    #pragma unroll
    for (int it = 0; it < 4; ++it) {
      int f4 = t + it * 256;
      int n  = f4 >> 3;
      int c0 = (f4 & 7) << 2;
      const float4 v = *(const float4*)(W + (size_t)n * DD + kp + c0);
      *(float4*)&Ws[n * XSTR + c0] = v;
    }
    __syncthreads();

    #pragma unroll
    for (int nt = 0; nt < 8; ++nt) {
      #pragma unroll
      for (int kk = 0; kk < 32; kk += 4) {
        v2f a  = *(const v2f*)&Xs[(m0 + lm) * XSTR + kk + 2 * hl];
        v2f bb = *(const v2f*)&Ws[(nt * 16 + lm) * XSTR + kk + 2 * hl];
        acc[nt] = wmma4(a, bb, acc[nt]);
      }
    }
  }

  #pragma unroll
  for (int nt = 0; nt < 8; ++nt)
    #pragma unroll
    for (int r = 0; r < 8; ++r)
      Y[(size_t)(rbase + m0 + r + 8 * hl) * DD + nt * 16 + lm] = acc[nt][r];
}

// ---------------------------------------------------------------------------
// Kernel 3: fused masked attention, flash-style (online softmax).
// Grid: (N/64, B*S*H). Block: 128 threads = 4 waves; wave owns 16 query rows.
// K stored row-major [key][feat] (QK^T B-frags contiguous); V stored
// transposed [feat][key] (P.V B-frags contiguous). All frags = ds_load_b64.
// ---------------------------------------------------------------------------
#define KSTR 20    // multiple of 4; 20*r mod 64 distinct for r<16
#define VSTR 132   // multiple of 4; 132*c mod 64 = 4c distinct for c<16
#define PSTR 132

__global__ __launch_bounds__(128) void attn_kernel(
    const float* __restrict__ qb, const float* __restrict__ kb,
    const float* __restrict__ vb, const float* __restrict__ bias,
    float* __restrict__ ob) {
  __shared__ float Ks[128 * KSTR];       // 10 KB   [key][feat]
  __shared__ float VsT[16 * VSTR];       // 8.25 KB [feat][key]
  __shared__ float Ps[4 * 16 * PSTR];    // 33 KB   per-wave P tiles

  const int t    = threadIdx.x;
  const int lane = t & 31;
  const int w    = t >> 5;
  const int hl   = lane >> 4;
  const int lm   = lane & 15;

  const int bsh = blockIdx.y;
  const int h   = bsh & (HH - 1);
  const int bs  = bsh >> 3;          // b*S + s
  const int b   = bs / SS;
  const size_t base = (size_t)bs * NN * DD + h * HD;
  const int qr0 = blockIdx.x * 64 + w * 16;

  // Q fragments for this wave's 16 rows (A layout), kept in registers.
  v2f qf[4];
  const float* Q = qb + base + (size_t)(qr0 + lm) * DD;
  #pragma unroll
  for (int kt = 0; kt < 4; ++kt)
    qf[kt] = *(const v2f*)(Q + 4 * kt + 2 * hl);

  float m_run[8], l_run[8];
  #pragma unroll
  for (int r = 0; r < 8; ++r) { m_run[r] = -__builtin_huge_valf(); l_run[r] = 0.0f; }
  v8f acc;
  #pragma unroll
  for (int r = 0; r < 8; ++r) acc[r] = 0.0f;

  const float scale = 0.25f;  // 1/sqrt(HD)
  float* Pw = &Ps[w * 16 * PSTR];
  const float* brows = bias + (size_t)b * NN * NN;

  for (int j0 = 0; j0 < NN; j0 += 128) {
    __syncthreads();
    // Stage K block row-major and V block transposed (512 float4 / 128 thr).
    #pragma unroll
    for (int it = 0; it < 4; ++it) {
      int f4  = t + it * 128;
      int row = f4 >> 2;
      int c0  = (f4 & 3) << 2;
      const float* kp = kb + base + (size_t)(j0 + row) * DD + c0;
      const float* vp = vb + base + (size_t)(j0 + row) * DD + c0;
      const float4 kv = *(const float4*)kp;
      const float4 vv = *(const float4*)vp;
      *(float4*)&Ks[row * KSTR + c0] = kv;
      VsT[(c0 + 0) * VSTR + row] = vv.x;
      VsT[(c0 + 1) * VSTR + row] = vv.y;
      VsT[(c0 + 2) * VSTR + row] = vv.z;
      VsT[(c0 + 3) * VSTR + row] = vv.w;
      if (c0 == 0 && j0 + 128 < NN) {
        __builtin_prefetch(kp + 128 * DD, 0, 0);   // next K block row
        __builtin_prefetch(vp + 128 * DD, 0, 0);   // next V block row
      }
    }
    __syncthreads();

    // Scores: 8 tiles of 16x16 = this wave's 16 q-rows x 128 keys.
    v8f st[8];
    #pragma unroll
    for (int nt = 0; nt < 8; ++nt) {
      v8f c;
      #pragma unroll
      for (int r = 0; r < 8; ++r) c[r] = 0.0f;
      #pragma unroll
      for (int kt = 0; kt < 4; ++kt) {
        v2f bbf = *(const v2f*)&Ks[(nt * 16 + lm) * KSTR + 4 * kt + 2 * hl];
        c = wmma4(qf[kt], bbf, c);
      }
      #pragma unroll
      for (int r = 0; r < 8; ++r) {
        int qg = qr0 + r + 8 * hl;
        int jg = j0 + nt * 16 + lm;
        st[nt][r] = c[r] * scale + brows[(size_t)qg * NN + jg];
      }
    }

    // Online softmax bookkeeping per row (rows live in 16-lane halves).
    #pragma unroll
    for (int r = 0; r < 8; ++r) {
      float mx = st[0][r];
      #pragma unroll
      for (int nt = 1; nt < 8; ++nt) mx = fmaxf(mx, st[nt][r]);
      #pragma unroll
      for (int off = 1; off < 16; off <<= 1)
        mx = fmaxf(mx, __shfl_xor(mx, off, 32));
      float mnew = fmaxf(m_run[r], mx);
      float cr = __expf(m_run[r] - mnew);
      m_run[r] = mnew;
      float rs = 0.0f;
      #pragma unroll
      for (int nt = 0; nt < 8; ++nt) {
        float p = __expf(st[nt][r] - mnew);
        st[nt][r] = p;
        rs += p;
      }
      #pragma unroll
      for (int off = 1; off < 16; off <<= 1) rs += __shfl_xor(rs, off, 32);
      l_run[r] = l_run[r] * cr + rs;
      acc[r] *= cr;
    }

    // Spill P (C layout) to LDS so it can be re-read in A layout.
    #pragma unroll
    for (int nt = 0; nt < 8; ++nt)
      #pragma unroll
      for (int r = 0; r < 8; ++r)
        Pw[(r + 8 * hl) * PSTR + nt * 16 + lm] = st[nt][r];
    __syncthreads();

    // acc += P(16x128) @ V(128x16); both fragments contiguous ds_load_b64.
    #pragma unroll
    for (int kk = 0; kk < 128; kk += 4) {
      v2f a   = *(const v2f*)&Pw[lm * PSTR + kk + 2 * hl];
      v2f bbf = *(const v2f*)&VsT[lm * VSTR + kk + 2 * hl];
      acc = wmma4(a, bbf, acc);
    }
  }

  #pragma unroll
  for (int r = 0; r < 8; ++r) {
    float inv = 1.0f / l_run[r];
    ob[base + (size_t)(qr0 + r + 8 * hl) * DD + lm] = acc[r] * inv;
  }
}

// ---------------------------------------------------------------------------
// Launcher
// ---------------------------------------------------------------------------
extern "C" void kernel_launch(void* const* d_in, const int* in_sizes, int n_in,
                              void* d_out, int out_size, void* d_ws, size_t ws_size,
                              hipStream_t stream) {
  const float* query = (const float*)d_in[0];
  const float* key   = (const float*)d_in[1];
  const float* value = (const float*)d_in[2];
  const float* x     = (const float*)d_in[3];
  const float* Wq    = (const float*)d_in[4];
  const float* bq    = (const float*)d_in[5];
  const float* Wk    = (const float*)d_in[6];
  const float* bk    = (const float*)d_in[7];
  const float* Wv    = (const float*)d_in[8];
  const float* bv    = (const float*)d_in[9];
  const float* Wo    = (const float*)d_in[10];
  const float* bo    = (const float*)d_in[11];

  float* ws = (float*)d_ws;
  const size_t per = (size_t)BB * SS * NN * DD;  // 3,145,728 floats
  float* qbuf = ws;
  float* kbuf = ws + per;
  float* vbuf = ws + 2 * per;
  float* obuf = ws + 3 * per;
  float* bias = ws + 4 * per;                    // B*N*N floats

  const int rowTiles = (BB * SS * NN) / 128;     // 192

  lsh_mask_kernel<<<dim3(BB * NN), 256, 0, stream>>>(x, bias);
  proj_kernel<<<dim3(rowTiles), 256, 0, stream>>>(query, Wq, bq, qbuf);
  proj_kernel<<<dim3(rowTiles), 256, 0, stream>>>(key,   Wk, bk, kbuf);
  proj_kernel<<<dim3(rowTiles), 256, 0, stream>>>(value, Wv, bv, vbuf);
  attn_kernel<<<dim3(NN / 64, BB * SS * HH), 128, 0, stream>>>(qbuf, kbuf, vbuf,
                                                               bias, obuf);
  proj_kernel<<<dim3(rowTiles), 256, 0, stream>>>(obuf, Wo, bo, (float*)d_out);
}